// DCRNN_79620103733874
// MI455X (gfx1250) — compile-verified
//
#include <hip/hip_runtime.h>
#include <cmath>
#include <cstdint>

// ---------------- problem constants (match reference) ----------------
#define Bdim   8
#define Tsteps 12
#define Ndim   2000
#define DINdim 2
#define Hdim   64
#define Edim   32000
#define HORdim 12
#define Mrows  (Bdim * Ndim)   // 16000 rows for all GEMMs

#define CEILDIV(a, b) (((a) + (b) - 1) / (b))

typedef float v2f __attribute__((ext_vector_type(2)));
typedef float v8f __attribute__((ext_vector_type(8)));

// Max padded-K across layers (384) + 4-float pad to decorrelate LDS banks
#define LDSTR_MAX 388

// ---------------------------------------------------------------------
// Copy weight [Nout, K] into zero-padded [Nout, Kp] workspace
// ---------------------------------------------------------------------
__global__ void pad_w_kernel(float* __restrict__ dst, const float* __restrict__ src,
                             int Nout, int K, int Kp) {
    int total = Nout * Kp;
    for (int i = blockIdx.x * blockDim.x + threadIdx.x; i < total;
         i += gridDim.x * blockDim.x) {
        int n = i / Kp, k = i - n * Kp;
        dst[i] = (k < K) ? src[(size_t)n * K + k] : 0.0f;
    }
}

// ---------------------------------------------------------------------
// xt[b*N+n, d] = source[b, t, n, d]
// ---------------------------------------------------------------------
__global__ void slice_src_kernel(float* __restrict__ xt, const float* __restrict__ source,
                                 int t) {
    int total = Mrows * DINdim;
    for (int i = blockIdx.x * blockDim.x + threadIdx.x; i < total;
         i += gridDim.x * blockDim.x) {
        int m = i / DINdim, d = i - m * DINdim;
        int b = m / Ndim, n = m - b * Ndim;
        xt[i] = source[(((size_t)b * Tsteps + t) * Ndim + n) * DINdim + d];
    }
}

// ---------------------------------------------------------------------
// XD[m, 0:Fin] = in[m,:]; XD[m, Fin:Fin+H] = h[m,:] (optionally * r-gate)
// gates layout: [M, 2H], r = gates[:, H:2H]
// ---------------------------------------------------------------------
__global__ void concat_kernel(float* __restrict__ XD, int ldx,
                              const float* __restrict__ in, int Fin,
                              const float* __restrict__ h,
                              const float* __restrict__ gates, int use_r) {
    int F = Fin + Hdim;
    int total = Mrows * F;
    for (int i = blockIdx.x * blockDim.x + threadIdx.x; i < total;
         i += gridDim.x * blockDim.x) {
        int m = i / F, f = i - m * F;
        float v;
        if (f < Fin) {
            v = in[(size_t)m * Fin + f];
        } else {
            int j = f - Fin;
            v = h[(size_t)m * Hdim + j];
            if (use_r) v *= gates[(size_t)m * (2 * Hdim) + Hdim + j];
        }
        XD[(size_t)m * ldx + f] = v;
    }
}

// ---------------------------------------------------------------------
// Scatter-add graph conv: dst[b, row[e], f] += w[e] * src[b, col[e], f]
// One block per edge; dst must be zero-initialized.
// ---------------------------------------------------------------------
__global__ __launch_bounds__(256)
void graph_conv_kernel(float* __restrict__ dst, int dst_off, int ld_dst,
                       const float* __restrict__ src, int src_off, int ld_src,
                       const int* __restrict__ rowi, const int* __restrict__ coli,
                       const float* __restrict__ w, int F) {
    int e = blockIdx.x;
    int r = rowi[e], c = coli[e];
    float we = w[e];
    int tot = Bdim * F;
    for (int i = threadIdx.x; i < tot; i += blockDim.x) {
        int b = i / F, f = i - b * F;
        float v = we * src[(size_t)(b * Ndim + c) * ld_src + src_off + f];
        atomicAdd(&dst[(size_t)(b * Ndim + r) * ld_dst + dst_off + f], v);
    }
}

// ---------------------------------------------------------------------
// Chebyshev T2: XD[m, 2F+f] = 2*Tb[m,f] - XD[m, f]
// ---------------------------------------------------------------------
__global__ void cheb2_kernel(float* __restrict__ XD, int ldx,
                             const float* __restrict__ Tb, int F) {
    int total = Mrows * F;
    for (int i = blockIdx.x * blockDim.x + threadIdx.x; i < total;
         i += gridDim.x * blockDim.x) {
        int m = i / F, f = i - m * F;
        XD[(size_t)m * ldx + 2 * F + f] =
            2.0f * Tb[(size_t)m * F + f] - XD[(size_t)m * ldx + f];
    }
}

// ---------------------------------------------------------------------
// WMMA fp32 GEMM with async-to-LDS A staging:
//   out[M, Nout] = act(A[M, Kp] @ W[Nout, Kp]^T + bias)
// grid.x = M/16 (one 16-row strip per block); blockDim = (Nout/16)*32.
// Block stages the 16 x Kp A-strip into padded LDS once via
// GLOBAL_LOAD_ASYNC_TO_LDS_B128 (ASYNCcnt), then each wave computes one
// 16x16 tile with V_WMMA_F32_16X16X4_F32, A from LDS, B from global (L2).
// ACT: 0 = none, 1 = sigmoid, 2 = tanh
// ---------------------------------------------------------------------
template <int ACT>
__global__ __launch_bounds__(256)
void gemm_wmma_lds_kernel(const float* __restrict__ A, const float* __restrict__ W,
                          const float* __restrict__ bias, float* __restrict__ out,
                          int Nout, int Kp, int lda, int ldo) {
    __shared__ __align__(16) float sA[16 * LDSTR_MAX];

    const int ldstr = Kp + 4;            // padded LDS row stride (floats)
    const int m0    = blockIdx.x << 4;
    const int tid   = threadIdx.x;

    // ---- stage A strip [16 x Kp] into LDS (16B async transfers) ----
    const int vecs_per_row = Kp >> 2;            // b128 chunks per row
    const int total_vecs   = vecs_per_row << 4;  // 16 rows
    for (int v = tid; v < total_vecs; v += blockDim.x) {
        int row = v / vecs_per_row;
        int cid = v - row * vecs_per_row;
        const float* gp = A + (size_t)(m0 + row) * lda + (cid << 2);
        unsigned ldsoff = (unsigned)(uintptr_t)(&sA[row * ldstr + (cid << 2)]);
        asm volatile("global_load_async_to_lds_b128 %0, %1, off"
                     :: "v"(ldsoff), "v"(gp)
                     : "memory");
    }
    asm volatile("s_wait_asynccnt 0x0" ::: "memory");
    __syncthreads();

    // ---- per-wave 16x16 tile ----
    const int wave = tid >> 5;
    const int lane = tid & 31;
    const int n0   = wave << 4;

    // ISA fp32 layouts:
    //  A 16x4 : lane L -> row M = L&15, K pair = ((L>>4)*2) + {0,1}
    //  B 4x16 : lane L -> col N = L&15, K pair = ((L>>4)*2) + {0,1}
    const int lr    = lane & 15;
    const int koff  = (lane >> 4) << 1;
    const int abase = lr * ldstr + koff;
    const float* brow = W + (size_t)(n0 + lr) * Kp + koff;

    v8f acc = {};
    for (int k = 0; k < Kp; k += 4) {
        v2f a = *(const v2f*)(&sA[abase + k]);   // ds_load_b64, conflict-free
        v2f b = *(const v2f*)(brow + k);         // global (weights live in L2)
        acc = __builtin_amdgcn_wmma_f32_16x16x4_f32(
            /*neg_a=*/false, a, /*neg_b=*/false, b,
            /*c_mod=*/(short)0, acc, /*reuse_a=*/false, /*reuse_b=*/false);
    }

    // C/D 16x16 f32: VGPR j, lanes 0-15 -> M=j, lanes 16-31 -> M=j+8; N = lane&15
    const int rbase = m0 + ((lane >> 4) << 3);
    const int cidx  = n0 + lr;
    const float bv  = bias[cidx];
#pragma unroll
    for (int j = 0; j < 8; ++j) {
        float v = acc[j] + bv;
        if (ACT == 1)      v = 1.0f / (1.0f + __expf(-v));
        else if (ACT == 2) v = tanhf(v);
        out[(size_t)(rbase + j) * ldo + cidx] = v;
    }
}

// ---------------------------------------------------------------------
// GRU update: h = (1-z)*h + z*cand;  z = G[:, 0:H]
// ---------------------------------------------------------------------
__global__ void update_h_kernel(float* __restrict__ h, const float* __restrict__ G,
                                const float* __restrict__ C) {
    int total = Mrows * Hdim;
    for (int i = blockIdx.x * blockDim.x + threadIdx.x; i < total;
         i += gridDim.x * blockDim.x) {
        int m = i / Hdim, j = i - m * Hdim;
        float z = G[(size_t)m * (2 * Hdim) + j];
        h[i] = (1.0f - z) * h[i] + z * C[i];
    }
}

// ---------------------------------------------------------------------
// Output projection: out[b, hor, n, 0] = dot(h1[b,n,:], Wout[hor,:]) + bout[hor]
// ---------------------------------------------------------------------
__global__ void out_proj_kernel(float* __restrict__ out, const float* __restrict__ h1,
                                const float* __restrict__ Wout,
                                const float* __restrict__ bout) {
    int total = Bdim * HORdim * Ndim;
    for (int i = blockIdx.x * blockDim.x + threadIdx.x; i < total;
         i += gridDim.x * blockDim.x) {
        int b   = i / (HORdim * Ndim);
        int rem = i - b * (HORdim * Ndim);
        int hor = rem / Ndim;
        int n   = rem - hor * Ndim;
        const float* hp = h1 + (size_t)(b * Ndim + n) * Hdim;
        const float* wp = Wout + (size_t)hor * Hdim;
        float s = bout[hor];
#pragma unroll 4
        for (int k = 0; k < Hdim; ++k) s += hp[k] * wp[k];
        out[i] = s;
    }
}

// ---------------------------------------------------------------------
// Host helper: one DCGRU cell
// ---------------------------------------------------------------------
static void run_cell(const float* in, int Fin, float* h,
                     const float* Wg, const float* bg,
                     const float* Wc, const float* bc, int ldx,
                     float* XD, float* Tb, float* G, float* Cb,
                     const int* rowi, const int* coli, const float* ew,
                     hipStream_t stream) {
    int F = Fin + Hdim;
    int ew_total = Mrows * F;

    // ---- gates path ----
    hipMemsetAsync(XD, 0, (size_t)Mrows * ldx * sizeof(float), stream);
    concat_kernel<<<CEILDIV(ew_total, 256), 256, 0, stream>>>(XD, ldx, in, Fin, h, G, 0);
    graph_conv_kernel<<<Edim, 256, 0, stream>>>(XD, F, ldx, XD, 0, ldx, rowi, coli, ew, F);
    hipMemsetAsync(Tb, 0, (size_t)Mrows * F * sizeof(float), stream);
    graph_conv_kernel<<<Edim, 256, 0, stream>>>(Tb, 0, F, XD, F, ldx, rowi, coli, ew, F);
    cheb2_kernel<<<CEILDIV(ew_total, 256), 256, 0, stream>>>(XD, ldx, Tb, F);

    // gates: Nout = 2H = 128 -> 8 waves/block, one 16-row strip per block
    gemm_wmma_lds_kernel<1><<<Mrows / 16, (2 * Hdim / 16) * 32, 0, stream>>>(
        XD, Wg, bg, G, 2 * Hdim, ldx, ldx, 2 * Hdim);

    // ---- candidate path ----
    hipMemsetAsync(XD, 0, (size_t)Mrows * ldx * sizeof(float), stream);
    concat_kernel<<<CEILDIV(ew_total, 256), 256, 0, stream>>>(XD, ldx, in, Fin, h, G, 1);
    graph_conv_kernel<<<Edim, 256, 0, stream>>>(XD, F, ldx, XD, 0, ldx, rowi, coli, ew, F);
    hipMemsetAsync(Tb, 0, (size_t)Mrows * F * sizeof(float), stream);
    graph_conv_kernel<<<Edim, 256, 0, stream>>>(Tb, 0, F, XD, F, ldx, rowi, coli, ew, F);
    cheb2_kernel<<<CEILDIV(ew_total, 256), 256, 0, stream>>>(XD, ldx, Tb, F);

    // candidate: Nout = H = 64 -> 4 waves/block
    gemm_wmma_lds_kernel<2><<<Mrows / 16, (Hdim / 16) * 32, 0, stream>>>(
        XD, Wc, bc, Cb, Hdim, ldx, ldx, Hdim);

    // ---- h update ----
    update_h_kernel<<<CEILDIV(Mrows * Hdim, 256), 256, 0, stream>>>(h, G, Cb);
}

// ---------------------------------------------------------------------
// Entry point
// ---------------------------------------------------------------------
extern "C" void kernel_launch(void* const* d_in, const int* in_sizes, int n_in,
                              void* d_out, int out_size, void* d_ws, size_t ws_size,
                              hipStream_t stream) {
    (void)in_sizes; (void)n_in; (void)out_size; (void)ws_size;

    const float* source = (const float*)d_in[0];
    const int*   eidx   = (const int*)d_in[1];
    const int*   rowi   = eidx;            // edge_index[0] = dst
    const int*   coli   = eidx + Edim;     // edge_index[1] = src
    const float* ew     = (const float*)d_in[2];
    const float* Wg0 = (const float*)d_in[3];
    const float* bg0 = (const float*)d_in[4];
    const float* Wc0 = (const float*)d_in[5];
    const float* bc0 = (const float*)d_in[6];
    const float* Wg1 = (const float*)d_in[7];
    const float* bg1 = (const float*)d_in[8];
    const float* Wc1 = (const float*)d_in[9];
    const float* bc1 = (const float*)d_in[10];
    const float* Wout = (const float*)d_in[11];
    const float* bout = (const float*)d_in[12];
    float* out = (float*)d_out;

    // ---- workspace layout (floats; all offsets multiples of 4 floats = 16B) ----
    const int K0 = 198, K0p = 200;   // layer0: 3*(2+64)   -> pad to /4
    const int K1 = 384, K1p = 384;   // layer1: 3*(64+64)
    float* ws = (float*)d_ws;
    size_t off = 0;
    float* Wg0p = ws + off; off += (size_t)128 * K0p;
    float* Wc0p = ws + off; off += (size_t)64  * K0p;
    float* Wg1p = ws + off; off += (size_t)128 * K1p;
    float* Wc1p = ws + off; off += (size_t)64  * K1p;
    float* h0   = ws + off; off += (size_t)Mrows * Hdim;
    float* h1   = ws + off; off += (size_t)Mrows * Hdim;
    float* xt   = ws + off; off += (size_t)Mrows * DINdim;
    float* XD   = ws + off; off += (size_t)Mrows * K1p;      // shared by both layers
    float* Tb   = ws + off; off += (size_t)Mrows * 128;
    float* G    = ws + off; off += (size_t)Mrows * 128;
    float* Cb   = ws + off; off += (size_t)Mrows * 64;

    // ---- prep: padded weights, zero hidden state ----
    pad_w_kernel<<<CEILDIV(128 * K0p, 256), 256, 0, stream>>>(Wg0p, Wg0, 128, K0, K0p);
    pad_w_kernel<<<CEILDIV(64  * K0p, 256), 256, 0, stream>>>(Wc0p, Wc0, 64,  K0, K0p);
    pad_w_kernel<<<CEILDIV(128 * K1p, 256), 256, 0, stream>>>(Wg1p, Wg1, 128, K1, K1p);
    pad_w_kernel<<<CEILDIV(64  * K1p, 256), 256, 0, stream>>>(Wc1p, Wc1, 64,  K1, K1p);
    hipMemsetAsync(h0, 0, (size_t)Mrows * Hdim * sizeof(float), stream);
    hipMemsetAsync(h1, 0, (size_t)Mrows * Hdim * sizeof(float), stream);

    // ---- recurrent time loop ----
    for (int t = 0; t < Tsteps; ++t) {
        slice_src_kernel<<<CEILDIV(Mrows * DINdim, 256), 256, 0, stream>>>(xt, source, t);
        run_cell(xt, DINdim, h0, Wg0p, bg0, Wc0p, bc0, K0p,
                 XD, Tb, G, Cb, rowi, coli, ew, stream);
        run_cell(h0, Hdim,  h1, Wg1p, bg1, Wc1p, bc1, K1p,
                 XD, Tb, G, Cb, rowi, coli, ew, stream);
    }

    // ---- output projection ----
    out_proj_kernel<<<CEILDIV(Bdim * HORdim * Ndim, 256), 256, 0, stream>>>(
        out, h1, Wout, bout);
}